// Casrel_42228118454862
// MI455X (gfx1250) — compile-verified
//
#include <hip/hip_runtime.h>
#include <hip/hip_bf16.h>

#define BB 32
#define SS 512
#define HH 1024
#define RR 64
#define NCOL 128          // Wo_h (64) || Wo_t (64)
#define LDSTR 1028        // padded LDS row stride (floats): 16 lanes * stride4 -> 64 distinct banks

typedef __attribute__((ext_vector_type(2))) float v2f;
typedef __attribute__((ext_vector_type(4))) float v4f;
typedef __attribute__((ext_vector_type(8))) float v8f;

__device__ __forceinline__ float bce_logits(float z, float y) {
    // matches F.binary_cross_entropy_with_logits (stable form)
    return fmaxf(z, 0.0f) - z * y + log1pf(__expf(-fabsf(z)));
}

// ---------------------------------------------------------------- accum init
__global__ void casrel_zero(double* acc) {
    if (threadIdx.x < 2) acc[threadIdx.x] = 0.0;
}

// -------------------------------------------------- weight repack (B frags)
// Wpack[((k0/4)*2 + half)*NCOL + gcol] = { Wcat[k0+2h][gcol], Wcat[k0+2h+1][gcol] }
// so each lane's B fragment is one contiguous 8-byte load (16 lanes -> 128 B),
// and consecutive K-steps advance the address by a constant 2048 B.
__global__ void casrel_pack_w(const float* __restrict__ WoH,
                              const float* __restrict__ WoT,
                              float* __restrict__ Wp) {
    const int idx = blockIdx.x * blockDim.x + threadIdx.x;   // 0 .. (H/4)*2*NCOL
    if (idx >= (HH / 4) * 2 * NCOL) return;
    const int gcol = idx % NCOL;
    const int rest = idx / NCOL;
    const int half = rest & 1;
    const int k    = (rest >> 1) * 4 + 2 * half;
    const float* W = (gcol < RR) ? (WoH + gcol) : (WoT + (gcol - RR));
    Wp[idx * 2 + 0] = W[(size_t)k * RR];
    Wp[idx * 2 + 1] = W[(size_t)(k + 1) * RR];
}

// ------------------------------------------------- subject pooling (sparse)
// subject[b,h] = 0.5*( sum_s head_w*ctx + sum_s tail_w*ctx ); weights one-hot.
// Also precompute subjWo[b,n] = subject[b,:] @ Wcat[:,n]  (the broadcast GEMM
// correction, per-batch constant, so the main kernel's LDS tile is raw ctx).
__global__ void casrel_subject_pool(const float* __restrict__ ctx,
                                    const float* __restrict__ wh,
                                    const float* __restrict__ wt,
                                    const float* __restrict__ WoH,
                                    const float* __restrict__ WoT,
                                    float* __restrict__ subjWo) {
    __shared__ int   hIdx[64]; __shared__ float hW[64]; __shared__ int hCnt;
    __shared__ int   tIdx[64]; __shared__ float tW[64]; __shared__ int tCnt;
    __shared__ float subj[HH];
    const int b = blockIdx.x;
    if (threadIdx.x == 0) { hCnt = 0; tCnt = 0; }
    __syncthreads();
    for (int s = threadIdx.x; s < SS; s += blockDim.x) {
        float vh = wh[b * SS + s];
        if (vh != 0.0f) { int i = atomicAdd(&hCnt, 1); if (i < 64) { hIdx[i] = s; hW[i] = vh; } }
        float vt = wt[b * SS + s];
        if (vt != 0.0f) { int i = atomicAdd(&tCnt, 1); if (i < 64) { tIdx[i] = s; tW[i] = vt; } }
    }
    __syncthreads();
    const int nh = min(hCnt, 64), nt = min(tCnt, 64);
    for (int h = threadIdx.x; h < HH; h += blockDim.x) {
        float acc = 0.0f;
        for (int i = 0; i < nh; ++i) acc += hW[i] * ctx[((size_t)b * SS + hIdx[i]) * HH + h];
        for (int i = 0; i < nt; ++i) acc += tW[i] * ctx[((size_t)b * SS + tIdx[i]) * HH + h];
        subj[h] = 0.5f * acc;
    }
    __syncthreads();
    for (int n = threadIdx.x; n < NCOL; n += blockDim.x) {
        const float* W = (n < RR) ? (WoH + n) : (WoT + (n - RR));
        float acc = 0.0f;
        for (int k = 0; k < HH; ++k) acc += subj[k] * W[(size_t)k * RR];
        subjWo[b * NCOL + n] = acc;
    }
}

// ------------------------------------------------------------- fused main
// One block = 16 consecutive tokens (never crosses a batch boundary: S%16==0).
// 128 threads = 4 waves. Waves 0,1 -> Wo_h columns, waves 2,3 -> Wo_t columns.
__global__ __launch_bounds__(128)
void casrel_main(const float* __restrict__ ctx,
                 const float* __restrict__ masks,
                 const float* __restrict__ goldSH,
                 const float* __restrict__ goldST,
                 const float* __restrict__ goldOH,
                 const float* __restrict__ goldOT,
                 const float* __restrict__ Wsh, const float* __restrict__ bsh,
                 const float* __restrict__ Wst, const float* __restrict__ bst,
                 const float* __restrict__ boH, const float* __restrict__ boT,
                 const float* __restrict__ Wp,       // packed B fragments
                 const float* __restrict__ subjWo,   // [B, NCOL]
                 double* __restrict__ acc) {
    __shared__ float tileX[16 * LDSTR];   // raw context tile, padded rows
    __shared__ float rowMask[16];

    const int t    = threadIdx.x;
    const int row0 = blockIdx.x * 16;        // global token row of tile
    const int b    = row0 / SS;

    float localLoss = 0.0f;
    float localMask = 0.0f;

    // ---- Phase 1: stage raw ctx into LDS; subject logits (VALU dots).
    {
        const int m = t >> 3;                // row within tile (0..15)
        const int j = t & 7;                 // 8 threads sweep one row
        const size_t rbase = (size_t)(row0 + m) * HH;
        const v4f* c4  = (const v4f*)(ctx + rbase);
        const v4f* wh4 = (const v4f*)Wsh;
        const v4f* wt4 = (const v4f*)Wst;
        float ah = 0.0f, at = 0.0f;
        #pragma unroll 4
        for (int it = 0; it < 32; ++it) {
            const int c = j + it * 8;        // float4 column index
            v4f v = c4[c];
            v4f a = wh4[c], bw = wt4[c];
            ah += v.x * a.x + v.y * a.y + v.z * a.z + v.w * a.w;
            at += v.x * bw.x + v.y * bw.y + v.z * bw.z + v.w * bw.w;
            *(v4f*)&tileX[m * LDSTR + 4 * c] = v;
        }
        // reduce over the 8 lanes of this row (contiguous lanes of one wave32)
        for (int d = 4; d > 0; d >>= 1) { ah += __shfl_down(ah, d, 8); at += __shfl_down(at, d, 8); }
        if (j == 0) {
            const int s = (row0 + m) % SS;
            const float mk = masks[b * SS + s];
            const float zh = ah + bsh[0];
            const float zt = at + bst[0];
            localLoss += (bce_logits(zh, goldSH[b * SS + s]) +
                          bce_logits(zt, goldST[b * SS + s])) * mk;
            localMask += mk;                 // each token row counted exactly once
            rowMask[m] = mk;
        }
    }
    __syncthreads();

    // ---- Phase 2: object GEMM via V_WMMA_F32_16X16X4_F32 on raw ctx.
    const int w    = t >> 5;                 // wave 0..3
    const int lane = t & 31;
    const int half = lane >> 4;              // K-half select (A layout: K = vgpr + 2*half)
    const int ln   = lane & 15;

    const int gcol0 = w * 32 + ln;           // global Wcat column of first N tile
    const int gcol1 = gcol0 + 16;            // second N tile

    v8f acc0 = {}; v8f acc1 = {};
    const float* aLds = &tileX[ln * LDSTR + 2 * half];
    // running base pointer: +512 floats (2048 B) per K-step -> immediate offsets
    const float* wpb = Wp + ((size_t)half * NCOL + (size_t)gcol0) * 2;
    #pragma unroll 8
    for (int k0 = 0; k0 < HH; k0 += 4, wpb += 512) {
        // A fragment: rows = lanes 0..15, K = k0 + 2*half + {0,1}
        v2f av  = *(const v2f*)(aLds + k0);
        // B fragments: one b64 per lane from the packed layout
        v2f bv0 = *(const v2f*)(wpb);
        v2f bv1 = *(const v2f*)(wpb + 32);
        acc0 = __builtin_amdgcn_wmma_f32_16x16x4_f32(false, av, false, bv0,
                                                     (short)0, acc0, false, false);
        acc1 = __builtin_amdgcn_wmma_f32_16x16x4_f32(false, av, false, bv1,
                                                     (short)0, acc1, false, false);
    }

    // ---- Phase 3: BCE epilogue. C/D layout: VGPR r -> M = r + 8*half, N = ln.
    const bool isHead = (w < 2);
    const float* bo = isHead ? boH : boT;
    const float* Gd = isHead ? goldOH : goldOT;
    const int col0 = gcol0 & (RR - 1);       // column within head/tail block
    const int col1 = gcol1 & (RR - 1);
    const float c0 = bo[col0] + subjWo[b * NCOL + gcol0];
    const float c1 = bo[col1] + subjWo[b * NCOL + gcol1];
    #pragma unroll
    for (int r = 0; r < 8; ++r) {
        const int m = r + 8 * half;
        const float mk = rowMask[m];
        const size_t gro = (size_t)(row0 + m) * RR;
        localLoss += bce_logits(acc0[r] + c0, Gd[gro + col0]) * mk;
        localLoss += bce_logits(acc1[r] + c1, Gd[gro + col1]) * mk;
    }

    // ---- wave reduce + one f64 atomic per wave
    float ls = localLoss, ms = localMask;
    for (int d = 16; d > 0; d >>= 1) { ls += __shfl_down(ls, d, 32); ms += __shfl_down(ms, d, 32); }
    if (lane == 0) {
        atomicAdd(&acc[0], (double)ls);
        if (ms != 0.0f) atomicAdd(&acc[1], (double)ms);
    }
}

__global__ void casrel_finalize(const double* __restrict__ acc, float* __restrict__ out) {
    out[0] = (float)(acc[0] / acc[1]);
}

extern "C" void kernel_launch(void* const* d_in, const int* in_sizes, int n_in,
                              void* d_out, int out_size, void* d_ws, size_t ws_size,
                              hipStream_t stream) {
    const float* ctx    = (const float*)d_in[0];
    const float* masks  = (const float*)d_in[1];
    const float* goldSH = (const float*)d_in[2];
    const float* goldST = (const float*)d_in[3];
    const float* subH   = (const float*)d_in[4];
    const float* subT   = (const float*)d_in[5];
    const float* goldOH = (const float*)d_in[6];
    const float* goldOT = (const float*)d_in[7];
    const float* Wsh = (const float*)d_in[8];
    const float* bsh = (const float*)d_in[9];
    const float* Wst = (const float*)d_in[10];
    const float* bst = (const float*)d_in[11];
    const float* WoH = (const float*)d_in[12];
    const float* boH = (const float*)d_in[13];
    const float* WoT = (const float*)d_in[14];
    const float* boT = (const float*)d_in[15];

    // workspace layout
    double* acc    = (double*)d_ws;                              // 2 doubles
    float*  subjWo = (float*)((char*)d_ws + 64);                 // B*NCOL floats (16 KB)
    float*  Wp     = (float*)((char*)d_ws + 64 + 16384);         // H*NCOL floats (512 KB)

    casrel_zero<<<1, 32, 0, stream>>>(acc);
    casrel_pack_w<<<((HH / 4) * 2 * NCOL + 255) / 256, 256, 0, stream>>>(WoH, WoT, Wp);
    casrel_subject_pool<<<BB, 256, 0, stream>>>(ctx, subH, subT, WoH, WoT, subjWo);
    casrel_main<<<(BB * SS) / 16, 128, 0, stream>>>(
        ctx, masks, goldSH, goldST, goldOH, goldOT,
        Wsh, bsh, Wst, bst, boH, boT, Wp, subjWo, acc);
    casrel_finalize<<<1, 1, 0, stream>>>(acc, (float*)d_out);
}